// SememeRGCN_50637664420137
// MI455X (gfx1250) — compile-verified
//
#include <hip/hip_runtime.h>

typedef __attribute__((ext_vector_type(2))) float v2f;
typedef __attribute__((ext_vector_type(8))) float v8f;
typedef __attribute__((address_space(3))) float lfloat_t;   // LDS float (32-bit ptr)

constexpr int NN = 50000;   // nodes
constexpr int EE = 800000;  // edges
constexpr int RR = 8;       // relations
constexpr int DD = 128;     // embed dim
constexpr int HH = 128;     // hidden dim
constexpr int KW = RR * DD;     // 1024  (relation-weight K extent)
constexpr int KT = KW + DD;     // 1152  (+ root K extent)
constexpr int LDA = 17;         // padded LDS stride for transposed A tiles
constexpr float EPSF = 1e-5f;

#if __has_builtin(__builtin_amdgcn_global_load_async_to_lds_b32) && \
    __has_builtin(__builtin_amdgcn_s_wait_asynccnt)
#define HAVE_ASYNC_LDS 1
typedef __attribute__((address_space(1))) int gint_t;   // global int (AS1)
typedef __attribute__((address_space(3))) int lint_t;   // LDS int (AS3)
#endif

// ---------------------------------------------------------------- clear ----
__global__ __launch_bounds__(256) void rgcn_clear(float* __restrict__ p, long n4) {
  long i = (long)blockIdx.x * blockDim.x + threadIdx.x;
  long stride = (long)gridDim.x * blockDim.x;
  float4* p4 = (float4*)p;
  float4 z = make_float4(0.f, 0.f, 0.f, 0.f);
  for (; i < n4; i += stride) p4[i] = z;
}

// --------------------------------------------------------------- gather ----
// x[i, :] = emb[node_ids[i], :]   (one wave32 per row, float4 per lane)
__global__ __launch_bounds__(256) void rgcn_gather(const float* __restrict__ emb,
                                                   const int* __restrict__ ids,
                                                   float* __restrict__ x) {
  int gid = blockIdx.x * 256 + threadIdx.x;
  int row = gid >> 5, lane = gid & 31;
  if (row >= NN) return;
  int src = ids[row];
  const float4* in = (const float4*)(emb + (size_t)src * DD);
  float4* out = (float4*)(x + (size_t)row * DD);
  out[lane] = in[lane];
}

// -------------------------------------------------------------- scatter ----
// agg[(dst*R + t), :] += x[src, :]; cnt[dst*R + t] += 1   (one wave per edge)
__global__ __launch_bounds__(256) void rgcn_scatter(const float* __restrict__ x,
                                                    const int* __restrict__ ei,
                                                    const int* __restrict__ et,
                                                    float* __restrict__ agg,
                                                    float* __restrict__ cnt) {
  int gid = blockIdx.x * 256 + threadIdx.x;
  int e = gid >> 5, lane = gid & 31;
  if (e >= EE) return;
  int src = ei[e];
  int dst = ei[EE + e];
  int t   = et[e];
  const float4* xv = (const float4*)(x + (size_t)src * DD);
  float4 v = xv[lane];
  float* dp = agg + ((size_t)dst * RR + t) * DD + lane * 4;
  __hip_atomic_fetch_add(dp + 0, v.x, __ATOMIC_RELAXED, __HIP_MEMORY_SCOPE_AGENT);
  __hip_atomic_fetch_add(dp + 1, v.y, __ATOMIC_RELAXED, __HIP_MEMORY_SCOPE_AGENT);
  __hip_atomic_fetch_add(dp + 2, v.z, __ATOMIC_RELAXED, __HIP_MEMORY_SCOPE_AGENT);
  __hip_atomic_fetch_add(dp + 3, v.w, __ATOMIC_RELAXED, __HIP_MEMORY_SCOPE_AGENT);
  if (lane == 0)
    __hip_atomic_fetch_add(cnt + (size_t)dst * RR + t, 1.0f,
                           __ATOMIC_RELAXED, __HIP_MEMORY_SCOPE_AGENT);
}

// ----------------------------------------------------------------- gemm ----
// Two 16-node row tiles per workgroup (M=32): each B fragment streamed from
// L2 is reused by both tiles -> halves weight traffic vs one-tile blocking.
// A tiles staged transposed in LDS ([k][row], stride 17, conflict-free for
// staging writes and both 16-lane fragment read groups). Tile base pointers
// are laundered through empty inline asm (as 32-bit AS3 pointers) so each
// tile keeps its own DS base register and all fragment pairs encode as
// ds_load_2addr with immediate offsets. 1/max(cnt,1) fused into staging;
// x-part staged via async global->LDS when the builtins exist.
__global__ __launch_bounds__(256) void rgcn_gemm(const float* __restrict__ agg,
                                                 const float* __restrict__ cnt,
                                                 const float* __restrict__ xin,
                                                 const float* __restrict__ W,
                                                 const float* __restrict__ root,
                                                 const float* __restrict__ bias,
                                                 float* __restrict__ out) {
  __shared__ float ldsA0[KT * LDA];   // tile 0: 78336 B
  __shared__ float ldsA1[KT * LDA];   // tile 1: 78336 B  (total ~157 KB of 320 KB)
  __shared__ float ldsInv[32 * RR];   // 1/max(cnt,1) per (row, relation)

  const int tid = threadIdx.x;
  const int node0 = blockIdx.x * 32;

  {  // 256 threads cover all 32*8 inv entries
    int row = tid >> 3, r = tid & (RR - 1);
    float c = (node0 + row < NN) ? cnt[(size_t)(node0 + row) * RR + r] : 1.0f;
    ldsInv[tid] = 1.0f / fmaxf(c, 1.0f);
  }

  // x-part of A: A[row][KW+d] = xin[node0+row][d]  (no scaling -> async DMA)
#if HAVE_ASYNC_LDS
  for (int idx = tid; idx < 32 * DD; idx += 256) {
    int row = idx >> 7;
    int d   = idx & (DD - 1);
    float* lp = ((row < 16) ? ldsA0 : ldsA1) + (KW + d) * LDA + (row & 15);
    const float* gp = xin + (size_t)(node0 + row) * DD + d;
    __builtin_amdgcn_global_load_async_to_lds_b32((gint_t*)gp, (lint_t*)lp, 0, 0);
  }
#endif

  __syncthreads();  // ldsInv visible

  // mean part: A[row][k] = agg[node0+row][k] * inv[row][k/128]
  for (int idx = tid; idx < 32 * KW; idx += 256) {
    int row = idx >> 10;            // / 1024
    int k   = idx & (KW - 1);
    float v = agg[(size_t)(node0 + row) * KW + k] * ldsInv[(row << 3) | (k >> 7)];
    float* base = (row < 16) ? ldsA0 : ldsA1;
    base[k * LDA + (row & 15)] = v;
  }

#if HAVE_ASYNC_LDS
  __builtin_amdgcn_s_wait_asynccnt(0);
#else
  for (int idx = tid; idx < 32 * DD; idx += 256) {
    int row = idx >> 7;
    int d   = idx & (DD - 1);
    float* base = (row < 16) ? ldsA0 : ldsA1;
    base[(KW + d) * LDA + (row & 15)] = xin[(size_t)(node0 + row) * DD + d];
  }
#endif
  __syncthreads();

  const int lane = tid & 31;
  const int wave = tid >> 5;
  const int hi = lane >> 4;        // 0/1: lane half selects K sub-rows & M half
  const int lm = lane & 15;        // A: M index, B/D: N index
  const int col = wave * 16 + lm;  // global output column
  const int aoff = 2 * hi * LDA + lm;

  v8f acc00 = {}, acc01 = {};      // tile 0, two K-chains
  v8f acc10 = {}, acc11 = {};      // tile 1, two K-chains

  // ---- segment 1: relation weights, k in [0, 1024) ----
  {
    const lfloat_t* a0p = (const lfloat_t*)(ldsA0 + aoff);
    const lfloat_t* a1p = (const lfloat_t*)(ldsA1 + aoff);
    asm volatile("" : "+v"(a0p));   // opaque: keep separate DS base registers
    asm volatile("" : "+v"(a1p));
    const float* bp  = W + (size_t)(2 * hi) * HH + col;
    #pragma unroll 2
    for (int it = 0; it < KW / 8; ++it) {
      v2f b0, b1, a00, a01, a10, a11;
      b0.x  = bp[0];            b0.y  = bp[HH];
      b1.x  = bp[4 * HH];       b1.y  = bp[5 * HH];
      a00.x = a0p[0];           a00.y = a0p[LDA];
      a01.x = a0p[4 * LDA];     a01.y = a0p[5 * LDA];
      a10.x = a1p[0];           a10.y = a1p[LDA];
      a11.x = a1p[4 * LDA];     a11.y = a1p[5 * LDA];
      acc00 = __builtin_amdgcn_wmma_f32_16x16x4_f32(false, a00, false, b0,
                                                    (short)0, acc00, false, false);
      acc10 = __builtin_amdgcn_wmma_f32_16x16x4_f32(false, a10, false, b0,
                                                    (short)0, acc10, false, false);
      acc01 = __builtin_amdgcn_wmma_f32_16x16x4_f32(false, a01, false, b1,
                                                    (short)0, acc01, false, false);
      acc11 = __builtin_amdgcn_wmma_f32_16x16x4_f32(false, a11, false, b1,
                                                    (short)0, acc11, false, false);
      a0p += 8 * LDA;
      a1p += 8 * LDA;
      bp  += (size_t)8 * HH;
    }
  }

  // ---- segment 2: root weights, k in [0, 128) ----
  {
    const lfloat_t* a0p = (const lfloat_t*)(ldsA0 + KW * LDA + aoff);
    const lfloat_t* a1p = (const lfloat_t*)(ldsA1 + KW * LDA + aoff);
    asm volatile("" : "+v"(a0p));
    asm volatile("" : "+v"(a1p));
    const float* bp  = root + (size_t)(2 * hi) * HH + col;
    #pragma unroll 2
    for (int it = 0; it < DD / 8; ++it) {
      v2f b0, b1, a00, a01, a10, a11;
      b0.x  = bp[0];            b0.y  = bp[HH];
      b1.x  = bp[4 * HH];       b1.y  = bp[5 * HH];
      a00.x = a0p[0];           a00.y = a0p[LDA];
      a01.x = a0p[4 * LDA];     a01.y = a0p[5 * LDA];
      a10.x = a1p[0];           a10.y = a1p[LDA];
      a11.x = a1p[4 * LDA];     a11.y = a1p[5 * LDA];
      acc00 = __builtin_amdgcn_wmma_f32_16x16x4_f32(false, a00, false, b0,
                                                    (short)0, acc00, false, false);
      acc10 = __builtin_amdgcn_wmma_f32_16x16x4_f32(false, a10, false, b0,
                                                    (short)0, acc10, false, false);
      acc01 = __builtin_amdgcn_wmma_f32_16x16x4_f32(false, a01, false, b1,
                                                    (short)0, acc01, false, false);
      acc11 = __builtin_amdgcn_wmma_f32_16x16x4_f32(false, a11, false, b1,
                                                    (short)0, acc11, false, false);
      a0p += 8 * LDA;
      a1p += 8 * LDA;
      bp  += (size_t)8 * HH;
    }
  }

  // ---- epilogue: bias + store (D layout: vgpr j -> row j + 8*hi, col = lm) ----
  float bv = bias[col];
  #pragma unroll
  for (int j = 0; j < 8; ++j) {
    int row0 = node0 + j + 8 * hi;           // tile 0 rows always < NN
    out[(size_t)row0 * HH + col] = acc00[j] + acc01[j] + bv;
    int row1 = row0 + 16;                    // tile 1 may run past N (last WG)
    if (row1 < NN)
      out[(size_t)row1 * HH + col] = acc10[j] + acc11[j] + bv;
  }
}

// -------------------------------------------------------- layernorm+relu ----
// One wave per row of 128; float4 per lane; xor-shuffle reduction.
__global__ __launch_bounds__(256) void rgcn_ln_relu(const float* __restrict__ y,
                                                    const float* __restrict__ g,
                                                    const float* __restrict__ be,
                                                    float* __restrict__ out) {
  int gid = blockIdx.x * 256 + threadIdx.x;
  int row = gid >> 5, lane = gid & 31;
  if (row >= NN) return;
  const float4* yp = (const float4*)(y + (size_t)row * HH);
  float4 v = yp[lane];
  float s1 = v.x + v.y + v.z + v.w;
  float s2 = v.x * v.x + v.y * v.y + v.z * v.z + v.w * v.w;
  #pragma unroll
  for (int m = 16; m >= 1; m >>= 1) {
    s1 += __shfl_xor(s1, m, 32);
    s2 += __shfl_xor(s2, m, 32);
  }
  float mu = s1 * (1.0f / HH);
  float var = s2 * (1.0f / HH) - mu * mu;
  float rstd = rsqrtf(var + EPSF);
  const float4* gp = (const float4*)(g + lane * 4);
  const float4* bp = (const float4*)(be + lane * 4);
  float4 gv = gp[0], bv = bp[0];
  float4 o;
  o.x = fmaxf((v.x - mu) * rstd * gv.x + bv.x, 0.0f);
  o.y = fmaxf((v.y - mu) * rstd * gv.y + bv.y, 0.0f);
  o.z = fmaxf((v.z - mu) * rstd * gv.z + bv.z, 0.0f);
  o.w = fmaxf((v.w - mu) * rstd * gv.w + bv.w, 0.0f);
  float4* op = (float4*)(out + (size_t)row * HH);
  op[lane] = o;
}

// ---------------------------------------------------------------- launch ----
extern "C" void kernel_launch(void* const* d_in, const int* in_sizes, int n_in,
                              void* d_out, int out_size, void* d_ws, size_t ws_size,
                              hipStream_t stream) {
  const int*   node_ids = (const int*)  d_in[0];
  const int*   ei       = (const int*)  d_in[1];
  const int*   et       = (const int*)  d_in[2];
  const float* emb      = (const float*)d_in[3];
  const float* W1 = (const float*)d_in[4];  const float* root1 = (const float*)d_in[5];
  const float* b1 = (const float*)d_in[6];  const float* g1    = (const float*)d_in[7];
  const float* be1= (const float*)d_in[8];
  const float* W2 = (const float*)d_in[9];  const float* root2 = (const float*)d_in[10];
  const float* b2 = (const float*)d_in[11]; const float* g2    = (const float*)d_in[12];
  const float* be2= (const float*)d_in[13];
  const float* W3 = (const float*)d_in[14]; const float* root3 = (const float*)d_in[15];
  const float* b3 = (const float*)d_in[16];
  float* outp = (float*)d_out;

  // workspace carve-up (floats): xa, xb, y, agg, cnt   (cnt adjacent to agg!)
  float* xa  = (float*)d_ws;
  float* xb  = xa  + (size_t)NN * DD;
  float* y   = xb  + (size_t)NN * DD;
  float* agg = y   + (size_t)NN * HH;
  float* cnt = agg + (size_t)NN * RR * DD;

  const long clr4 = ((long)NN * RR * DD + (long)NN * RR) / 4;  // agg + cnt as float4
  const int rowBlocks  = (NN * 32) / 256;   // 6250
  const int edgeBlocks = (EE * 32) / 256;   // 100000
  const int gemmBlocks = (NN + 31) / 32;    // 1563 (last WG: tile 1 store-guarded)

  rgcn_gather<<<rowBlocks, 256, 0, stream>>>(emb, node_ids, xa);

  // ---- layer 1: xa -> y -> xb ----
  rgcn_clear  <<<8192, 256, 0, stream>>>(agg, clr4);
  rgcn_scatter<<<edgeBlocks, 256, 0, stream>>>(xa, ei, et, agg, cnt);
  rgcn_gemm   <<<gemmBlocks, 256, 0, stream>>>(agg, cnt, xa, W1, root1, b1, y);
  rgcn_ln_relu<<<rowBlocks, 256, 0, stream>>>(y, g1, be1, xb);

  // ---- layer 2: xb -> y -> xa ----
  rgcn_clear  <<<8192, 256, 0, stream>>>(agg, clr4);
  rgcn_scatter<<<edgeBlocks, 256, 0, stream>>>(xb, ei, et, agg, cnt);
  rgcn_gemm   <<<gemmBlocks, 256, 0, stream>>>(agg, cnt, xb, W2, root2, b2, y);
  rgcn_ln_relu<<<rowBlocks, 256, 0, stream>>>(y, g2, be2, xa);

  // ---- layer 3: xa -> d_out ----
  rgcn_clear  <<<8192, 256, 0, stream>>>(agg, clr4);
  rgcn_scatter<<<edgeBlocks, 256, 0, stream>>>(xa, ei, et, agg, cnt);
  rgcn_gemm   <<<gemmBlocks, 256, 0, stream>>>(agg, cnt, xa, W3, root3, b3, outp);
}